// PatchGINEncoder_42142219108934
// MI455X (gfx1250) — compile-verified
//
#include <hip/hip_runtime.h>
#include <hip/hip_bf16.h>

typedef __attribute__((ext_vector_type(2))) float v2f;
typedef __attribute__((ext_vector_type(8))) float v8f;

#define HDIM 128
#define NGRAPH 256
#define WPB 8    // waves per block in the fused MLP kernel
#define WPB_P 4  // waves per block in the projection kernel

// ---------------------------------------------------------------------------
// Wave-level GEMM: A = 16x128 strip (LDS, row-major), B = 128x128 row-major
// (LDS or global; addrspace inferred per call site). 8 acc tiles of 16x16.
// V_WMMA_F32_16X16X4_F32 fragment layout per CDNA5 ISA 7.12.2:
//   A: lane<16 -> row M=lane, K=kb+{0,1}; lane>=16 -> M=lane-16, K=kb+{2,3}
//   B/C/D: row striped across lanes within a VGPR.
// ---------------------------------------------------------------------------
__device__ __forceinline__ void wave_gemm_16x128(const float* As,
                                                 const float* B,
                                                 v8f acc[8], int lane) {
  const int mrow = lane & 15;
  const int ksel = (lane >> 4) << 1;  // 0 for lanes 0-15, 2 for lanes 16-31
  for (int kb = 0; kb < HDIM; kb += 4) {
    const int kq = kb + ksel;
    v2f a;
    a.x = As[mrow * HDIM + kq];
    a.y = As[mrow * HDIM + kq + 1];
    const float* B0 = B + kq * HDIM + mrow;
#pragma unroll
    for (int nt = 0; nt < 8; ++nt) {
      v2f b;
      b.x = B0[nt * 16];          // W[kq][col]
      b.y = B0[nt * 16 + HDIM];   // W[kq+1][col]  (512B apart -> 2addr LDS)
      acc[nt] = __builtin_amdgcn_wmma_f32_16x16x4_f32(
          false, a, false, b, (short)0, acc[nt], false, false);
    }
  }
}

// ---------------------------------------------------------------------------
// Simple utility kernels
// ---------------------------------------------------------------------------
__global__ void copy4_kernel(const float4* __restrict__ src,
                             float4* __restrict__ dst, int n4) {
  int i = blockIdx.x * blockDim.x + threadIdx.x;
  if (i < n4) dst[i] = src[i];
}

__global__ void zero_kernel(float* __restrict__ p, int n) {
  int i = blockIdx.x * blockDim.x + threadIdx.x;
  if (i < n) p[i] = 0.0f;
}

// ---------------------------------------------------------------------------
// Edge scatter: one wave per edge; lane handles 4 contiguous channels.
// z[dst] += h[src]  (z pre-initialized to h, giving (1+eps)*h + sum)
// ---------------------------------------------------------------------------
__global__ void scatter_kernel(const float* __restrict__ H,
                               const int* __restrict__ ei,
                               float* __restrict__ Z, int E) {
  const int lane = threadIdx.x & 31;
  const long long tid = (long long)blockIdx.x * blockDim.x + threadIdx.x;
  const int e = (int)(tid >> 5);
  if (e >= E) return;
  const int s = ei[e];
  const int d = ei[E + e];
  float4 v = ((const float4*)(H + (long long)s * HDIM))[lane];
  float* zp = Z + (long long)d * HDIM + lane * 4;
  unsafeAtomicAdd(zp + 0, v.x);
  unsafeAtomicAdd(zp + 1, v.y);
  unsafeAtomicAdd(zp + 2, v.z);
  unsafeAtomicAdd(zp + 3, v.w);
}

// ---------------------------------------------------------------------------
// Fused GIN MLP layer: h_out = ReLU( ReLU(BN(Z@W1+b1)) @ W2 + b2 )
// 8 waves/block; W1 (then W2) staged once per block into a shared 64KB LDS
// buffer; each wave owns an 8KB LDS A slice (Z strip, then intermediate).
// Dynamic LDS: 64KB Wbuf + 8*8KB A = 128KB (of 320KB/WGP).
// ---------------------------------------------------------------------------
__global__ void gin_mlp_kernel(const float* __restrict__ Z,
                               float* __restrict__ Hout,
                               const float* __restrict__ W1,
                               const float* __restrict__ b1,
                               const float* __restrict__ bng,
                               const float* __restrict__ bnb,
                               const float* __restrict__ bnm,
                               const float* __restrict__ bnv,
                               const float* __restrict__ W2,
                               const float* __restrict__ b2, int nRows) {
  extern __shared__ float smem[];            // [16384 Wbuf][WPB*2048 A]
  float* Wbuf = smem;
  float* myA = smem + HDIM * HDIM + (threadIdx.x >> 5) * (16 * HDIM);
  const int lane = threadIdx.x & 31;
  const int wave = threadIdx.x >> 5;
  const int rowBase = (blockIdx.x * WPB + wave) * 16;
  const bool valid = rowBase < nRows;  // wave-uniform (nRows % 16 == 0)

  // Cooperative stage of W1 into LDS (16384 floats across the block)
  {
    const float4* src = (const float4*)W1;
    float4* dst = (float4*)Wbuf;
    for (int i = threadIdx.x; i < HDIM * HDIM / 4; i += WPB * 32)
      dst[i] = src[i];
  }
  // Stage this wave's 16x128 Z strip into its LDS slice
  if (valid) {
    const float4* zsrc = (const float4*)(Z + (long long)rowBase * HDIM);
    float4* adst = (float4*)myA;
#pragma unroll
    for (int i = 0; i < 16; ++i) adst[i * 32 + lane] = zsrc[i * 32 + lane];
  }
  __syncthreads();

  const int mrow = lane & 15;
  const int hsel = lane >> 4;

  if (valid) {
    // GEMM1: Z @ W1 (A and B both from LDS)
    v8f acc[8];
#pragma unroll
    for (int i = 0; i < 8; ++i) acc[i] = (v8f)(0.0f);
    wave_gemm_16x128(myA, Wbuf, acc, lane);

    // Epilogue1: +b1, BatchNorm(eval), ReLU -> back into LDS slice
#pragma unroll
    for (int nt = 0; nt < 8; ++nt) {
      const int col = nt * 16 + mrow;
      const float s = bng[col] * rsqrtf(bnv[col] + 1e-5f);
      const float t = bnb[col] - bnm[col] * s;
      const float bias = b1[col];
#pragma unroll
      for (int r = 0; r < 8; ++r) {
        float v = acc[nt][r] + bias;
        v = v * s + t;
        v = fmaxf(v, 0.0f);
        myA[(r + 8 * hsel) * HDIM + col] = v;
      }
    }
  }
  __syncthreads();  // all waves done reading W1 from Wbuf

  // Cooperative stage of W2 into LDS
  {
    const float4* src = (const float4*)W2;
    float4* dst = (float4*)Wbuf;
    for (int i = threadIdx.x; i < HDIM * HDIM / 4; i += WPB * 32)
      dst[i] = src[i];
  }
  __syncthreads();

  if (valid) {
    // GEMM2: intermediate @ W2
    v8f acc2[8];
#pragma unroll
    for (int i = 0; i < 8; ++i) acc2[i] = (v8f)(0.0f);
    wave_gemm_16x128(myA, Wbuf, acc2, lane);

    // Epilogue2: +b2, ReLU -> global
#pragma unroll
    for (int nt = 0; nt < 8; ++nt) {
      const int col = nt * 16 + mrow;
      const float bias = b2[col];
#pragma unroll
      for (int r = 0; r < 8; ++r) {
        float v = fmaxf(acc2[nt][r] + bias, 0.0f);
        Hout[(long long)(rowBase + r + 8 * hsel) * HDIM + col] = v;
      }
    }
  }
}

// ---------------------------------------------------------------------------
// Mean pool: accumulate per-graph sums + counts with hardware f32 atomics
// ---------------------------------------------------------------------------
__global__ void pool_accum_kernel(const float* __restrict__ H,
                                  const int* __restrict__ batch,
                                  float* __restrict__ pool,
                                  float* __restrict__ cnt, int nNodes) {
  const int lane = threadIdx.x & 31;
  const long long tid = (long long)blockIdx.x * blockDim.x + threadIdx.x;
  const int node = (int)(tid >> 5);
  if (node >= nNodes) return;
  const int g = batch[node];
  float4 v = ((const float4*)(H + (long long)node * HDIM))[lane];
  float* pg = pool + (long long)g * HDIM + lane * 4;
  unsafeAtomicAdd(pg + 0, v.x);
  unsafeAtomicAdd(pg + 1, v.y);
  unsafeAtomicAdd(pg + 2, v.z);
  unsafeAtomicAdd(pg + 3, v.w);
  if (lane == 0) unsafeAtomicAdd(cnt + g, 1.0f);
}

__global__ void pool_finalize_kernel(const float* __restrict__ pool,
                                     const float* __restrict__ cnt,
                                     float* __restrict__ hg) {
  int i = blockIdx.x * blockDim.x + threadIdx.x;
  if (i < NGRAPH * HDIM) hg[i] = pool[i] / fmaxf(cnt[i >> 7], 1.0f);
}

// ---------------------------------------------------------------------------
// Projection GEMM: P = hg @ Wp + bp (256x128 @ 128x128), WMMA; Wp from global
// (only 16 waves total — staging would not amortize)
// ---------------------------------------------------------------------------
__global__ void proj_kernel(const float* __restrict__ HG,
                            const float* __restrict__ Wp,
                            const float* __restrict__ bp,
                            float* __restrict__ P) {
  __shared__ float As[WPB_P][16 * HDIM];
  const int lane = threadIdx.x & 31;
  const int wave = threadIdx.x >> 5;
  const int rowBase = (blockIdx.x * WPB_P + wave) * 16;
  if (rowBase >= NGRAPH) return;
  float* myA = As[wave];
  const float4* src = (const float4*)(HG + (long long)rowBase * HDIM);
  float4* adst = (float4*)myA;
#pragma unroll
  for (int i = 0; i < 16; ++i) adst[i * 32 + lane] = src[i * 32 + lane];

  v8f acc[8];
#pragma unroll
  for (int i = 0; i < 8; ++i) acc[i] = (v8f)(0.0f);
  wave_gemm_16x128(myA, Wp, acc, lane);

  const int mrow = lane & 15;
  const int hsel = lane >> 4;
#pragma unroll
  for (int nt = 0; nt < 8; ++nt) {
    const int col = nt * 16 + mrow;
    const float bias = bp[col];
#pragma unroll
    for (int r = 0; r < 8; ++r)
      P[(long long)(rowBase + r + 8 * hsel) * HDIM + col] = acc[nt][r] + bias;
  }
}

// ---------------------------------------------------------------------------
// LayerNorm over last dim (128): one wave per row, shfl_xor tree reduction
// ---------------------------------------------------------------------------
__global__ void ln_kernel(const float* __restrict__ P,
                          const float* __restrict__ g,
                          const float* __restrict__ b,
                          float* __restrict__ out) {
  const int lane = threadIdx.x & 31;
  const int row = (blockIdx.x * blockDim.x + threadIdx.x) >> 5;
  if (row >= NGRAPH) return;
  float4 v = ((const float4*)(P + (long long)row * HDIM))[lane];
  float s = v.x + v.y + v.z + v.w;
  float q = v.x * v.x + v.y * v.y + v.z * v.z + v.w * v.w;
#pragma unroll
  for (int off = 16; off >= 1; off >>= 1) {
    s += __shfl_xor(s, off, 32);
    q += __shfl_xor(q, off, 32);
  }
  const float mu = s * (1.0f / HDIM);
  const float var = q * (1.0f / HDIM) - mu * mu;
  const float inv = rsqrtf(var + 1e-5f);
  const int c0 = lane * 4;
  float4 o;
  o.x = (v.x - mu) * inv * g[c0 + 0] + b[c0 + 0];
  o.y = (v.y - mu) * inv * g[c0 + 1] + b[c0 + 1];
  o.z = (v.z - mu) * inv * g[c0 + 2] + b[c0 + 2];
  o.w = (v.w - mu) * inv * g[c0 + 3] + b[c0 + 3];
  ((float4*)(out + (long long)row * HDIM))[lane] = o;
}

// ---------------------------------------------------------------------------
// Host side
// ---------------------------------------------------------------------------
extern "C" void kernel_launch(void* const* d_in, const int* in_sizes, int n_in,
                              void* d_out, int out_size, void* d_ws,
                              size_t ws_size, hipStream_t stream) {
  const float* x    = (const float*)d_in[0];
  const int*   ei   = (const int*)d_in[1];
  const int*   batch= (const int*)d_in[2];
  const float* W1s  = (const float*)d_in[3];
  const float* b1s  = (const float*)d_in[4];
  const float* bng  = (const float*)d_in[5];
  const float* bnb  = (const float*)d_in[6];
  const float* bnm  = (const float*)d_in[7];
  const float* bnv  = (const float*)d_in[8];
  const float* W2s  = (const float*)d_in[9];
  const float* b2s  = (const float*)d_in[10];
  const float* Wp   = (const float*)d_in[11];
  const float* bp   = (const float*)d_in[12];
  const float* lng  = (const float*)d_in[13];
  const float* lnb  = (const float*)d_in[14];
  float* out = (float*)d_out;

  const int N = in_sizes[2];
  const int E = in_sizes[1] / 2;

  float* ws   = (float*)d_ws;
  float* h    = ws;                          // N*128
  float* z    = h + (size_t)N * HDIM;        // N*128
  float* pool = z + (size_t)N * HDIM;        // 256*128
  float* cnt  = pool + NGRAPH * HDIM;        // 256   (contiguous w/ pool)
  float* hg   = cnt + NGRAPH;                // 256*128
  float* p    = hg + NGRAPH * HDIM;          // 256*128

  const int n4 = N * HDIM / 4;
  const int copyBlocks = (n4 + 255) / 256;
  const int scatBlocks = (int)(((long long)E * 32 + 255) / 256);
  const int mlpBlocks  = (N / 16 + WPB - 1) / WPB;
  const size_t mlpLds  = (size_t)(HDIM * HDIM + WPB * 16 * HDIM) * sizeof(float);

  // h = x
  copy4_kernel<<<copyBlocks, 256, 0, stream>>>((const float4*)x, (float4*)h, n4);

  for (int l = 0; l < 3; ++l) {
    // z = h  (self term, eps = 0)
    copy4_kernel<<<copyBlocks, 256, 0, stream>>>((const float4*)h, (float4*)z, n4);
    // z[dst] += h[src]
    scatter_kernel<<<scatBlocks, 256, 0, stream>>>(h, ei, z, E);
    // h = ReLU(ReLU(BN(z@W1+b1))@W2 + b2)
    gin_mlp_kernel<<<mlpBlocks, 32 * WPB, mlpLds, stream>>>(
        z, h, W1s + (size_t)l * HDIM * HDIM, b1s + l * HDIM, bng + l * HDIM,
        bnb + l * HDIM, bnm + l * HDIM, bnv + l * HDIM,
        W2s + (size_t)l * HDIM * HDIM, b2s + l * HDIM, N);
  }

  // Mean pool per graph
  zero_kernel<<<(NGRAPH * HDIM + NGRAPH + 255) / 256, 256, 0, stream>>>(
      pool, NGRAPH * HDIM + NGRAPH);
  pool_accum_kernel<<<(int)(((long long)N * 32 + 255) / 256), 256, 0, stream>>>(
      h, batch, pool, cnt, N);
  pool_finalize_kernel<<<(NGRAPH * HDIM + 255) / 256, 256, 0, stream>>>(pool, cnt, hg);

  // Projection + LayerNorm
  proj_kernel<<<(NGRAPH / 16 + WPB_P - 1) / WPB_P, 32 * WPB_P, 0, stream>>>(hg, Wp, bp, p);
  ln_kernel<<<(NGRAPH * 32 + 255) / 256, 256, 0, stream>>>(p, lng, lnb, out);
}